// KarplusSynth_5514738008515
// MI455X (gfx1250) — compile-verified
//
#include <hip/hip_runtime.h>
#include <math.h>

// ---------------------------------------------------------------------------
// KarplusSynth for MI455X (gfx1250, wave32).
//  Kernel 1: MLP via V_WMMA_F32_16X16X4_F32 (exact f32 WMMA) + biquad setup.
//  Kernel 2: sequential Karplus-Strong scan, 441-float state in LDS,
//            one barrier/step, fused fade + ASR envelope, prefetch of next chunk.
// ---------------------------------------------------------------------------

typedef __attribute__((ext_vector_type(2))) float v2f;
typedef __attribute__((ext_vector_type(8))) float v8f;

#define WSZ 441
#define SR_F 44100.0f

// ---------------------------------------------------------------------------
// Kernel 1: one wave (32 threads).
//   hidden = relu(h @ W1 + b1)        [1,128]   (8 tiles x 3 K-steps of WMMA)
//   lat    = relu(hidden @ W2 + b2)   [1,5]     (32 K-steps of WMMA)
//   scalars -> ws_scalars; wavetable (2 biquads) -> ws_wt
// ---------------------------------------------------------------------------
__global__ __launch_bounds__(32)
void ks_setup_kernel(const float* __restrict__ h,
                     const float* __restrict__ W1,
                     const float* __restrict__ b1v,
                     const float* __restrict__ W2,
                     const float* __restrict__ b2v,
                     const float* __restrict__ lp_cutoff,
                     const float* __restrict__ wnoise,
                     float* __restrict__ ws_scalars,
                     float* __restrict__ ws_wt)
{
    __shared__ float hid[128];
    __shared__ float lat[8];

    const int lane = threadIdx.x;      // 0..31, full wave (EXEC all ones at WMMA)
    const int m    = lane & 15;        // row / column index within tile
    const int kh   = lane >> 4;        // which K-half this lane holds

    // ---- layer 1: h[1,9] @ W1[9,128] ------------------------------------
    for (int nt = 0; nt < 8; ++nt) {
        v8f c = {};
        for (int kc = 0; kc < 3; ++kc) {
            const int k0 = kc * 4 + 2 * kh;   // A lanes 0-15: K=0,1 ; 16-31: K=2,3
            const int k1 = k0 + 1;
            v2f a, b;
            a.x = (m == 0 && k0 < 9) ? h[k0] : 0.0f;
            a.y = (m == 0 && k1 < 9) ? h[k1] : 0.0f;
            b.x = (k0 < 9) ? W1[k0 * 128 + nt * 16 + m] : 0.0f;
            b.y = (k1 < 9) ? W1[k1 * 128 + nt * 16 + m] : 0.0f;
            c = __builtin_amdgcn_wmma_f32_16x16x4_f32(
                    false, a, false, b, (short)0, c, false, false);
        }
        if (lane < 16) {                      // D row M=0: c[0], lanes 0..15
            float v = c[0] + b1v[nt * 16 + lane];
            hid[nt * 16 + lane] = v > 0.0f ? v : 0.0f;
        }
    }
    __syncthreads();

    // ---- layer 2: hidden[1,128] @ W2[128,5] (N padded to 16) ------------
    v8f c2 = {};
    for (int kc = 0; kc < 32; ++kc) {
        const int k0 = kc * 4 + 2 * kh;
        v2f a, b;
        a.x = (m == 0) ? hid[k0]     : 0.0f;
        a.y = (m == 0) ? hid[k0 + 1] : 0.0f;
        b.x = (m < 5) ? W2[k0 * 5 + m]       : 0.0f;
        b.y = (m < 5) ? W2[(k0 + 1) * 5 + m] : 0.0f;
        c2 = __builtin_amdgcn_wmma_f32_16x16x4_f32(
                false, a, false, b, (short)0, c2, false, false);
    }
    if (lane < 5) {
        float v = c2[0] + b2v[lane];
        lat[lane] = v > 0.0f ? v : 0.0f;
    }
    __syncthreads();

    // ---- scalars + sequential biquads (one lane) ------------------------
    if (lane == 0) {
        const float l0 = lat[0], l1 = lat[1], l2 = lat[2], l3 = lat[3], l4 = lat[4];
        const float decay = fminf(fmaxf(l0 * 0.1f + 0.9f, 0.9f), 0.999f);
        const float lpf   = fminf(fmaxf(l1 * (SR_F * 0.25f), 100.0f), SR_F * 0.5f - 1.0f);
        const float lpq   = fminf(fmaxf(l2, 0.1f), 0.999f);
        ws_scalars[0] = decay;
        ws_scalars[1] = l3;   // feedback amount
        ws_scalars[2] = l4;   // output amplitude

        const float PI_F = 3.14159265358979323846f;

        // biquad #1: learned cutoff / Q, direct form II transposed
        {
            float w0 = 2.0f * PI_F * lpf / SR_F;
            float cw = cosf(w0);
            float al = sinf(w0) / (2.0f * lpq);
            float a0 = 1.0f + al;
            float cb0 = ((1.0f - cw) * 0.5f) / a0;
            float cb1 = (1.0f - cw) / a0;
            float cb2 = cb0;
            float ca1 = (-2.0f * cw) / a0;
            float ca2 = (1.0f - al) / a0;
            float s1 = 0.0f, s2 = 0.0f;
            for (int i = 0; i < WSZ; ++i) {
                float x = wnoise[i];
                float y = cb0 * x + s1;
                s1 = cb1 * x - ca1 * y + s2;
                s2 = cb2 * x - ca2 * y;
                ws_wt[i] = y;
            }
        }
        // biquad #2: fixed lowpass (lp_cutoff, Q=0.707), in place
        {
            float f  = lp_cutoff[0];
            float w0 = 2.0f * PI_F * f / SR_F;
            float cw = cosf(w0);
            float al = sinf(w0) / (2.0f * 0.707f);
            float a0 = 1.0f + al;
            float cb0 = ((1.0f - cw) * 0.5f) / a0;
            float cb1 = (1.0f - cw) / a0;
            float cb2 = cb0;
            float ca1 = (-2.0f * cw) / a0;
            float ca2 = (1.0f - al) / a0;
            float s1 = 0.0f, s2 = 0.0f;
            for (int i = 0; i < WSZ; ++i) {
                float x = ws_wt[i];
                float y = cb0 * x + s1;
                s1 = cb1 * x - ca1 * y + s2;
                s2 = cb2 * x - ca2 * y;
                ws_wt[i] = y;
            }
        }
    }
}

// ---------------------------------------------------------------------------
// Kernel 2: Karplus-Strong scan. One block, 448 threads (14 wave32s).
// State double-buffered in LDS; one barrier per chunk. Envelope fused.
// ---------------------------------------------------------------------------
__global__ __launch_bounds__(448)
void ks_main_kernel(const float* __restrict__ fb_line,
                    const float* __restrict__ env_params,
                    const float* __restrict__ t_arr,
                    const float* __restrict__ fade,
                    const float* __restrict__ ws_scalars,
                    const float* __restrict__ ws_wt,
                    float* __restrict__ out,
                    int n_samples, int n_chunks)
{
    __shared__ float buf[2][WSZ];

    const int tid = threadIdx.x;

    const float decay = ws_scalars[0];
    const float fa    = ws_scalars[1];
    const float amp   = ws_scalars[2];
    const float d2    = decay * 0.5f;

    const float ea = fabsf(env_params[0]) + 1e-3f;   // attack
    const float es = env_params[1];                   // sustain
    const float er = fabsf(env_params[2]) + 1e-3f;   // release
    const float T  = t_arr[n_samples - 1];
    const float inv_a = 1.0f / ea;
    const float inv_r = 1.0f / er;
    const float gain  = es * amp;
    const int fade_start = n_samples - 256;

    if (tid < WSZ) buf[0][tid] = ws_wt[tid];
    __syncthreads();

    int p = 0;
    for (int ck = 0; ck < n_chunks; ++ck) {
        const int base = ck * WSZ;
        if (tid < WSZ) {
            const int im1 = (tid == 0) ? (WSZ - 1) : (tid - 1);
            const float xi = buf[p][tid] + fa * fb_line[base + tid];
            const float xm = buf[p][im1] + fa * fb_line[base + im1];
            const float y  = d2 * (xi + xm);
            buf[p ^ 1][tid] = y;

            const int n = base + tid;
            const float tn = (float)n / SR_F;            // matches arange(N)/SR in f32
            float e1 = tn * inv_a;        e1 = fminf(fmaxf(e1, 0.0f), 1.0f);
            float e2 = (T - tn) * inv_r;  e2 = fminf(fmaxf(e2, 0.0f), 1.0f);
            float o = y;
            const int j = n - fade_start;
            if (j >= 0) o *= fade[j];
            out[n] = o * e1 * e2 * gain;

            // warm L2/L0 for the next chunk while we sit in the barrier
            if (ck + 1 < n_chunks)
                __builtin_prefetch(&fb_line[base + WSZ + tid], 0, 1);
        }
        __syncthreads();
        p ^= 1;
    }

    // Generic tail (reference handles rem>0; rem==0 for N=4,410,000).
    const int rem = n_samples - n_chunks * WSZ;
    if (rem > 0 && tid < rem) {
        const int im1 = (tid == 0) ? (WSZ - 1) : (tid - 1);
        const float y = d2 * (buf[p][tid] + buf[p][im1]);
        const int n = n_chunks * WSZ + tid;
        const float tn = (float)n / SR_F;
        float e1 = tn * inv_a;        e1 = fminf(fmaxf(e1, 0.0f), 1.0f);
        float e2 = (T - tn) * inv_r;  e2 = fminf(fmaxf(e2, 0.0f), 1.0f);
        float o = y;
        const int j = n - fade_start;
        if (j >= 0) o *= fade[j];
        out[n] = o * e1 * e2 * gain;
    }
}

// ---------------------------------------------------------------------------
extern "C" void kernel_launch(void* const* d_in, const int* in_sizes, int n_in,
                              void* d_out, int out_size, void* d_ws, size_t ws_size,
                              hipStream_t stream) {
    const float* fb          = (const float*)d_in[0];   // feedback_line [N]
    const float* h           = (const float*)d_in[1];   // [1,9]
    const float* t           = (const float*)d_in[2];   // [N]
    const float* wnoise      = (const float*)d_in[3];   // [441]
    const float* W1          = (const float*)d_in[4];   // [9,128]
    const float* b1v         = (const float*)d_in[5];   // [128]
    const float* W2          = (const float*)d_in[6];   // [128,5]
    const float* b2v         = (const float*)d_in[7];   // [5]
    const float* lp_cutoff   = (const float*)d_in[8];   // [1]
    const float* env_params  = (const float*)d_in[9];   // [3]
    const float* fade        = (const float*)d_in[10];  // [256]
    // d_in[11] = freq_rad, d_in[12] = output_length_samples (static; wsz = 441)

    float* ws_scalars = (float*)d_ws;        // [0..7]
    float* ws_wt      = (float*)d_ws + 8;    // [8..448]

    const int N        = out_size;
    const int n_chunks = N / WSZ;

    ks_setup_kernel<<<1, 32, 0, stream>>>(h, W1, b1v, W2, b2v, lp_cutoff,
                                          wnoise, ws_scalars, ws_wt);
    ks_main_kernel<<<1, 448, 0, stream>>>(fb, env_params, t, fade,
                                          ws_scalars, ws_wt,
                                          (float*)d_out, N, n_chunks);
}